// auxiva_29025388987151
// MI455X (gfx1250) — compile-verified
//
#include <hip/hip_runtime.h>
#include <stdint.h>

// ---------------------------------------------------------------------------
// AuxIVA on MI455X (gfx1250, wave32).
// K=4 sources, F=513 bins, B=4000 frames, 5 IP iterations.
// The B-contraction (phi-weighted Gram for V) runs on V_WMMA_F32_16X16X4_F32;
// everything else (demix, norms, 4x4 complex inverses) is tiny VALU work.
// All data stays fp32: the iterated 4x4 complex inverses are too sensitive
// for fp16/fp8 matrix paths. Entire working set (~66 MB) is L2-resident.
// ---------------------------------------------------------------------------

#define KSRC 4
#define FREQ 513
#define NB   4000
#define NITER 5

typedef __attribute__((ext_vector_type(2))) float v2f;
typedef __attribute__((ext_vector_type(8))) float v8f;

// ----------------------------- complex helpers -----------------------------
__device__ __forceinline__ float2 cadd(float2 a, float2 b){ return make_float2(a.x+b.x, a.y+b.y); }
__device__ __forceinline__ float2 csub(float2 a, float2 b){ return make_float2(a.x-b.x, a.y-b.y); }
__device__ __forceinline__ float2 cmul(float2 a, float2 b){ return make_float2(a.x*b.x - a.y*b.y, a.x*b.y + a.y*b.x); }
__device__ __forceinline__ float2 conj2(float2 a){ return make_float2(a.x, -a.y); }
__device__ __forceinline__ float  mag2(float2 a){ return a.x*a.x + a.y*a.y; }
__device__ __forceinline__ float2 cinv(float2 a){ float d = a.x*a.x + a.y*a.y; return make_float2(a.x/d, -a.y/d); }
__device__ __forceinline__ float2 csqrt2(float2 z){
  float m  = sqrtf(z.x*z.x + z.y*z.y);
  float re = sqrtf(fmaxf(0.5f*(m + z.x), 0.0f));
  float im = sqrtf(fmaxf(0.5f*(m - z.x), 0.0f));
  if (z.y < 0.0f) im = -im;
  return make_float2(re, im);
}

// Solve 4x4 complex A x = e_rhs via Gauss-Jordan with partial pivoting.
// (Matches jnp.linalg.inv column extraction.) A is clobbered.
__device__ void csolve4(float2 A[4][4], float2 x[4], int rhs){
  float2 b[4];
  #pragma unroll
  for (int i = 0; i < 4; ++i) b[i] = make_float2(i == rhs ? 1.0f : 0.0f, 0.0f);
  for (int col = 0; col < 4; ++col){
    int piv = col; float best = mag2(A[col][col]);
    for (int r = col + 1; r < 4; ++r){
      float m2 = mag2(A[r][col]);
      if (m2 > best){ best = m2; piv = r; }
    }
    if (piv != col){
      for (int j = 0; j < 4; ++j){ float2 t = A[col][j]; A[col][j] = A[piv][j]; A[piv][j] = t; }
      float2 t = b[col]; b[col] = b[piv]; b[piv] = t;
    }
    float2 ip = cinv(A[col][col]);
    for (int j = 0; j < 4; ++j) A[col][j] = cmul(A[col][j], ip);
    b[col] = cmul(b[col], ip);
    for (int r = 0; r < 4; ++r){
      if (r == col) continue;
      float2 fac = A[r][col];
      for (int j = 0; j < 4; ++j) A[r][j] = csub(A[r][j], cmul(fac, A[col][j]));
      b[r] = csub(b[r], cmul(fac, b[col]));
    }
  }
  for (int i = 0; i < 4; ++i) x[i] = b[i];
}

// ------------------------- kernel 0: Wh = identity -------------------------
__global__ void init_wh_kernel(float2* __restrict__ Wh){
  int idx = blockIdx.x * blockDim.x + threadIdx.x;
  if (idx >= FREQ * 16) return;
  int i = idx & 15;
  Wh[idx] = make_float2(((i >> 2) == (i & 3)) ? 1.0f : 0.0f, 0.0f);
}

// --------- kernel A: demix, accumulate |Y|^2 over f, emit phi[k][b] --------
__global__ void demix_phi_kernel(const float* __restrict__ Xr, const float* __restrict__ Xi,
                                 const float2* __restrict__ Wh, float* __restrict__ phi){
  int b = blockIdx.x * blockDim.x + threadIdx.x;
  if (b >= NB) return;
  const size_t FB = (size_t)FREQ * NB;
  float r2[KSRC] = {0.f, 0.f, 0.f, 0.f};
  for (int f = 0; f < FREQ; ++f){
    const float2* Wf = Wh + (size_t)f * 16;   // uniform per f-iteration
    float xr[KSRC], xi[KSRC];
    #pragma unroll
    for (int c = 0; c < KSRC; ++c){
      size_t o = (size_t)c * FB + (size_t)f * NB + b;
      xr[c] = Xr[o]; xi[c] = Xi[o];
    }
    #pragma unroll
    for (int s = 0; s < KSRC; ++s){
      float yr = 0.f, yi = 0.f;
      #pragma unroll
      for (int c = 0; c < KSRC; ++c){
        float2 w = Wf[s * 4 + c];
        yr += w.x * xr[c] - w.y * xi[c];
        yi += w.x * xi[c] + w.y * xr[c];
      }
      r2[s] += yr * yr + yi * yi;
    }
  }
  #pragma unroll
  for (int s = 0; s < KSRC; ++s)
    phi[(size_t)s * NB + b] = 0.5f / sqrtf(r2[s]);   // 1/(2*||Y_s||)
}

// ------ kernel C: V[k,i,j,f] = (1/B) sum_b phi[k,b] X[i,f,b] conj(X[j,f,b])
// WMMA f32 16x16x4: A rows 0-7 = phi_{k0}*u, rows 8-15 = phi_{k1}*u;
// B columns hold u (duplicated); contraction dim = 4 frames per instruction.
__global__ void __launch_bounds__(128)
gram_v_kernel(const float* __restrict__ Xr, const float* __restrict__ Xi,
              const float* __restrict__ phi, float2* __restrict__ V){
  const int f    = blockIdx.x;
  const int tid  = threadIdx.x;
  const int wave = tid >> 5;
  const int lane = tid & 31;
  const int half = lane >> 4;        // selects K (b-offset) group for A/B
  const int p    = lane & 7;         // u-row index (0..3 = Re ch, 4..7 = Im ch)
  const int ksel = (lane >> 3) & 1;  // which k within the packed pair

  const size_t FB = (size_t)FREQ * NB;
  const float* Urow = (p < 4 ? Xr + (size_t)p * FB : Xi + (size_t)(p - 4) * FB)
                      + (size_t)f * NB;
  const float* phiLo = phi + (size_t)(ksel)     * NB;   // pair {0,1}
  const float* phiHi = phi + (size_t)(2 + ksel) * NB;   // pair {2,3}

  v8f c0 = {};  // G for k=0 (rows 0-7) and k=1 (rows 8-15)
  v8f c1 = {};  // G for k=2 / k=3
  const int bbase = wave * (NB / 4);           // 4 waves x 1000 frames
  for (int s = 0; s < NB / 16; ++s){           // 250 steps of 4 frames
    const int bA = bbase + 4 * s + 2 * half;   // this lane's two b samples
    float2 u  = *(const float2*)(Urow  + bA);
    float2 pa = *(const float2*)(phiLo + bA);
    float2 pb = *(const float2*)(phiHi + bA);
    v2f a0; a0.x = u.x * pa.x; a0.y = u.y * pa.y;
    v2f a1; a1.x = u.x * pb.x; a1.y = u.y * pb.y;
    v2f bm; bm.x = u.x;        bm.y = u.y;
    c0 = __builtin_amdgcn_wmma_f32_16x16x4_f32(false, a0, false, bm, (short)0, c0, false, false);
    c1 = __builtin_amdgcn_wmma_f32_16x16x4_f32(false, a1, false, bm, (short)0, c1, false, false);
  }

  // reduce the 4 waves' partial Gram matrices in LDS
  __shared__ float Gred[2][16][16];
  for (int i = tid; i < 2 * 16 * 16; i += 128) ((float*)Gred)[i] = 0.0f;
  __syncthreads();
  const int Mbase = half * 8;      // C/D layout: lanes>=16 hold M = r+8
  const int Ncol  = lane & 15;
  #pragma unroll
  for (int r = 0; r < 8; ++r){
    atomicAdd(&Gred[0][Mbase + r][Ncol], c0[r]);
    atomicAdd(&Gred[1][Mbase + r][Ncol], c1[r]);
  }
  __syncthreads();

  if (tid < 64){
    int k = tid >> 4, i = (tid >> 2) & 3, j = tid & 3;
    int acc = k >> 1, off = (k & 1) * 8;
    float re, im;
    if (acc == 0){
      re = Gred[0][off + i][j]     + Gred[0][off + i + 4][j + 4];
      im = Gred[0][off + i + 4][j] - Gred[0][off + i][j + 4];
    } else {
      re = Gred[1][off + i][j]     + Gred[1][off + i + 4][j + 4];
      im = Gred[1][off + i + 4][j] - Gred[1][off + i][j + 4];
    }
    const float invB = 1.0f / (float)NB;
    V[((size_t)k * FREQ + f) * 16 + i * 4 + j] = make_float2(re * invB, im * invB);
  }
}

// -------- kernel D: sequential IP updates of Wh, one thread per f ----------
__global__ void ip_update_kernel(float2* __restrict__ Wh, const float2* __restrict__ V){
  int f = blockIdx.x * blockDim.x + threadIdx.x;
  if (f >= FREQ) return;
  float2 W[4][4];
  #pragma unroll
  for (int i = 0; i < 16; ++i) ((float2*)W)[i] = Wh[(size_t)f * 16 + i];
  for (int k = 0; k < KSRC; ++k){
    float2 Vk[4][4];
    const float2* vs = V + ((size_t)k * FREQ + f) * 16;
    #pragma unroll
    for (int i = 0; i < 16; ++i) ((float2*)Vk)[i] = vs[i];
    // M = W @ Vk
    float2 M[4][4];
    #pragma unroll
    for (int i = 0; i < 4; ++i)
      #pragma unroll
      for (int j = 0; j < 4; ++j){
        float2 s = make_float2(0.f, 0.f);
        #pragma unroll
        for (int c = 0; c < 4; ++c) s = cadd(s, cmul(W[i][c], Vk[c][j]));
        M[i][j] = s;
      }
    float2 x[4]; csolve4(M, x, k);              // inv(M)[:,k]
    float2 w[4];
    #pragma unroll
    for (int c = 0; c < 4; ++c) w[c] = conj2(x[c]);
    float2 s = make_float2(0.f, 0.f);           // w^T Vk conj(w)
    #pragma unroll
    for (int c = 0; c < 4; ++c)
      #pragma unroll
      for (int d = 0; d < 4; ++d)
        s = cadd(s, cmul(cmul(w[c], Vk[c][d]), conj2(w[d])));
    float2 idn = cinv(csqrt2(s));
    #pragma unroll
    for (int c = 0; c < 4; ++c) W[k][c] = cmul(w[c], idn);
  }
  #pragma unroll
  for (int i = 0; i < 16; ++i) Wh[(size_t)f * 16 + i] = ((float2*)W)[i];
}

// -------- kernel E: projection-back scale[f,s] = inv(Wh[f])[s,0] -----------
__global__ void scale_kernel(const float2* __restrict__ Wh, float2* __restrict__ scale){
  int f = blockIdx.x * blockDim.x + threadIdx.x;
  if (f >= FREQ) return;
  float2 A[4][4];
  #pragma unroll
  for (int i = 0; i < 16; ++i) ((float2*)A)[i] = Wh[(size_t)f * 16 + i];
  float2 x[4]; csolve4(A, x, 0);
  #pragma unroll
  for (int s = 0; s < 4; ++s) scale[(size_t)f * 4 + s] = x[s];
}

// -------- kernel F: out[s,f,b] = (Wh[f] X[:,f,b])_s * scale[f,s] -----------
__global__ void output_kernel(const float* __restrict__ Xr, const float* __restrict__ Xi,
                              const float2* __restrict__ Wh, const float2* __restrict__ scale,
                              float2* __restrict__ out){
  int b = blockIdx.x * blockDim.x + threadIdx.x;
  int f = blockIdx.y;
  if (b >= NB) return;
  const size_t FB = (size_t)FREQ * NB;
  const float2* Wf = Wh    + (size_t)f * 16;
  const float2* sc = scale + (size_t)f * 4;
  float xr[KSRC], xi[KSRC];
  #pragma unroll
  for (int c = 0; c < KSRC; ++c){
    size_t o = (size_t)c * FB + (size_t)f * NB + b;
    xr[c] = Xr[o]; xi[c] = Xi[o];
  }
  #pragma unroll
  for (int s = 0; s < KSRC; ++s){
    float yr = 0.f, yi = 0.f;
    #pragma unroll
    for (int c = 0; c < KSRC; ++c){
      float2 w = Wf[s * 4 + c];
      yr += w.x * xr[c] - w.y * xi[c];
      yi += w.x * xi[c] + w.y * xr[c];
    }
    out[((size_t)s * FREQ + f) * NB + b] = cmul(make_float2(yr, yi), sc[s]);
  }
}

// ---------------------------------------------------------------------------
extern "C" void kernel_launch(void* const* d_in, const int* in_sizes, int n_in,
                              void* d_out, int out_size, void* d_ws, size_t ws_size,
                              hipStream_t stream){
  const float* Xr = (const float*)d_in[0];
  const float* Xi = (const float*)d_in[1];

  uint8_t* p = (uint8_t*)d_ws;
  float2* Wh = (float2*)p;  p += (size_t)FREQ * 16 * sizeof(float2);        // 65.7 KB
  float2* V  = (float2*)p;  p += (size_t)KSRC * FREQ * 16 * sizeof(float2); // 262.7 KB
  float*  phi = (float*)p;  p += (size_t)KSRC * NB * sizeof(float);         // 64 KB
  float2* scale = (float2*)p;                                               // 16.4 KB
  float2* out = (float2*)d_out;   // complex64 (K,F,B) = interleaved float pairs

  init_wh_kernel<<<(FREQ * 16 + 255) / 256, 256, 0, stream>>>(Wh);
  for (int it = 0; it < NITER; ++it){
    demix_phi_kernel<<<(NB + 255) / 256, 256, 0, stream>>>(Xr, Xi, Wh, phi);
    gram_v_kernel<<<FREQ, 128, 0, stream>>>(Xr, Xi, phi, V);
    ip_update_kernel<<<(FREQ + 127) / 128, 128, 0, stream>>>(Wh, V);
  }
  scale_kernel<<<(FREQ + 127) / 128, 128, 0, stream>>>(Wh, scale);
  output_kernel<<<dim3((NB + 255) / 256, FREQ), 256, 0, stream>>>(Xr, Xi, Wh, scale, out);
}